// LitePTBackbone_67963562492329
// MI455X (gfx1250) — compile-verified
//
#include <hip/hip_runtime.h>
#include <stdint.h>

// ---------------- problem constants ----------------
#define GRIDSZ 0.02f
#define OUTC   72       // out channels
#define INC    6        // feat channels
#define NTILES 5        // 5*16 = 80 >= 72 output cols

// ---------------- hash table ----------------
#define HBITS  22
#define HSIZE  (1u << HBITS)      // 4.19M slots, load factor ~0.48 for 2M keys
#define HMASK  (HSIZE - 1u)
#define EMPTYK 0xFFFFFFFFu

typedef __attribute__((ext_vector_type(2))) float v2f;
typedef __attribute__((ext_vector_type(4))) float v4f;
typedef __attribute__((ext_vector_type(8))) float v8f;

__device__ __forceinline__ uint32_t mix32(uint32_t x) {
    x ^= x >> 16; x *= 0x7feb352dU;
    x ^= x >> 15; x *= 0x846ca68bU;
    x ^= x >> 16;
    return x;
}

// ---------------- pass 1: init hash table ----------------
__global__ void init_table(uint32_t* __restrict__ tkey, int* __restrict__ tmin) {
    uint32_t i = blockIdx.x * blockDim.x + threadIdx.x;
    if (i < HSIZE) {
        tkey[i] = EMPTYK;
        tmin[i] = 0x7FFFFFFF;
    }
}

// ---------------- pass 2: voxel key -> slot, atomicMin of point id ----------------
__global__ void build_table(const float* __restrict__ coord,
                            uint32_t* __restrict__ tkey,
                            int* __restrict__ tmin,
                            uint32_t* __restrict__ slot,
                            int n) {
    int i = blockIdx.x * blockDim.x + threadIdx.x;
    if (i >= n) return;
    // same f32 floor-divide as the reference -> identical voxel partition
    int gx = (int)floorf(coord[i * 3 + 0] / GRIDSZ);
    int gy = (int)floorf(coord[i * 3 + 1] / GRIDSZ);
    int gz = (int)floorf(coord[i * 3 + 2] / GRIDSZ);
    // coords in [0,10) => grid in [0,500): 10-bit fields are injective
    uint32_t key = ((uint32_t)gx << 20) | ((uint32_t)gy << 10) | (uint32_t)gz;
    uint32_t h = mix32(key) & HMASK;
    for (;;) {
        uint32_t old = atomicCAS(&tkey[h], EMPTYK, key);
        if (old == EMPTYK || old == key) {   // slot is write-once: unique per key
            atomicMin(&tmin[h], i);          // representative = min point id
            slot[i] = h;
            break;
        }
        h = (h + 1) & HMASK;
    }
}

// ---------------- pass 3: WMMA head + densify ----------------
__device__ __forceinline__ float loadA(const float* __restrict__ fr,
                                       const float* __restrict__ cr, int k) {
    if (k < INC)     return fr[k];
    if (k < INC + 3) return cr[k - INC];
    return 0.0f;
}

__device__ __forceinline__ float loadB(const float* __restrict__ W,
                                       const float* __restrict__ Wc,
                                       int k, int ncol) {
    if (ncol >= OUTC) return 0.0f;
    if (k < INC)      return W[k * OUTC + ncol];
    if (k < INC + 3)  return Wc[(k - INC) * OUTC + ncol];
    return 0.0f;
}

__global__ __launch_bounds__(256)
void voxel_head_wmma(const float* __restrict__ coord,
                     const float* __restrict__ feat,
                     const float* __restrict__ W,
                     const float* __restrict__ Wc,
                     const int* __restrict__ tmin,
                     const uint32_t* __restrict__ slot,
                     float* __restrict__ out,
                     int n) {
    // 8 waves/block, 16 points/wave; each wave owns a private 16x72 LDS tile.
    __shared__ __align__(16) float tile[8][16 * OUTC];   // 36.9 KB / block

    const int lane = threadIdx.x & 31;
    const int wave = threadIdx.x >> 5;
    const int half = lane >> 4;          // 0: K={4t,4t+1}  1: K={4t+2,4t+3}
    const int m    = lane & 15;
    const int base = (blockIdx.x * 8 + wave) * 16;   // 16 points per wave
    float* __restrict__ myTile = &tile[wave][0];

    // --- B fragments: [12 x 80] tile of (W ; Wc ; 0), 5 N-tiles x 3 K-tiles ---
    v2f Bf[NTILES][3];
#pragma unroll
    for (int j = 0; j < NTILES; ++j) {
        const int ncol = j * 16 + m;
#pragma unroll
        for (int t = 0; t < 3; ++t) {
            const int k0 = 4 * t + 2 * half;
            Bf[j][t].x = loadB(W, Wc, k0,     ncol);
            Bf[j][t].y = loadB(W, Wc, k0 + 1, ncol);
        }
    }

    // --- A fragments: 16 points x 12 K (feat | coord | pad), per-lane v2f x 3 ---
    int p = base + m;
    if (p >= n) p = n - 1;                   // N % 128 == 0 here, but stay safe
    const int rep = tmin[slot[p]];           // representative point of p's voxel
    const float* fr = feat  + (size_t)rep * INC;
    const float* cr = coord + (size_t)rep * 3;
    v2f Af[3];
#pragma unroll
    for (int t = 0; t < 3; ++t) {
        const int k0 = 4 * t + 2 * half;
        Af[t].x = loadA(fr, cr, k0);
        Af[t].y = loadA(fr, cr, k0 + 1);
    }

    // --- D = A x B on the matrix pipe, spill each 16x16 tile to LDS ---
#pragma unroll
    for (int j = 0; j < NTILES; ++j) {
        v8f acc = {};
#pragma unroll
        for (int t = 0; t < 3; ++t) {
            acc = __builtin_amdgcn_wmma_f32_16x16x4_f32(
                /*neg_a=*/false, Af[t],
                /*neg_b=*/false, Bf[j][t],
                /*c_mod=*/(short)0, acc,
                /*reuse_a=*/false, /*reuse_b=*/false);
        }
        const int ncol = j * 16 + m;
        // ncol < 72 is compile-time true for j < 4; only j==4 needs the lane guard
        if (j < 4 || m < 8) {
#pragma unroll
            for (int r = 0; r < 8; ++r) {
                myTile[(8 * half + r) * OUTC + ncol] = acc[r];
            }
        }
    }

    // --- drain: wave tile = contiguous 16*72 floats of `out`; coalesced b128 NT ---
    // (wave-private LDS region: in-wave DS ordering suffices, no barrier)
    const size_t outBase = (size_t)base * OUTC;
    if (base + 16 <= n) {
#pragma unroll
        for (int i = 0; i < 9; ++i) {           // 16*72/32 = 36 floats = 9 x v4f per lane
            const int idx = i * 128 + lane * 4; // 16B-aligned, wave-contiguous
            v4f v = *(const v4f*)&myTile[idx];
            __builtin_nontemporal_store(v, (v4f*)&out[outBase + idx]);
        }
    } else {
        for (int idx = lane; idx < 16 * OUTC; idx += 32) {
            if (base + idx / OUTC < n)
                __builtin_nontemporal_store(myTile[idx], &out[outBase + idx]);
        }
    }
}

// ---------------- launcher ----------------
extern "C" void kernel_launch(void* const* d_in, const int* in_sizes, int n_in,
                              void* d_out, int out_size, void* d_ws, size_t ws_size,
                              hipStream_t stream) {
    const float* coord = (const float*)d_in[0];   // [N,3]
    const float* feat  = (const float*)d_in[1];   // [N,6]
    const float* W     = (const float*)d_in[2];   // [6,72]
    const float* Wc    = (const float*)d_in[3];   // [3,72]
    float* out = (float*)d_out;                   // [N,72]
    const int n = in_sizes[0] / 3;

    // workspace layout: keys[H] | minidx[H] | slot[N]  (~41.6 MB)
    uint32_t* tkey = (uint32_t*)d_ws;
    int*      tmin = (int*)(tkey + HSIZE);
    uint32_t* slot = (uint32_t*)(tmin + HSIZE);

    init_table<<<(HSIZE + 255) / 256, 256, 0, stream>>>(tkey, tmin);
    build_table<<<(n + 255) / 256, 256, 0, stream>>>(coord, tkey, tmin, slot, n);
    voxel_head_wmma<<<(n + 127) / 128, 256, 0, stream>>>(coord, feat, W, Wc,
                                                         tmin, slot, out, n);
}